// Masker_84662395339312
// MI455X (gfx1250) — compile-verified
//
#include <hip/hip_runtime.h>
#include <hip/hip_bf16.h>

typedef __attribute__((ext_vector_type(2))) float v2f;
typedef __attribute__((ext_vector_type(8))) float v8f;

#define MTOT   64          // batch rows (fixed)
#define NTILE  64          // output columns per block
#define KC     32          // K chunk staged in LDS
#define LDSS   (KC + 2)    // padded LDS row stride (floats, even -> 8B-aligned v2f)
#define BN_EPS 1e-5f

// ---------------------------------------------------------------------------
// Fused GEMM (out = A @ W^T) + training-mode BatchNorm over the 64-row batch
// (+ optional affine gamma/beta, + optional ReLU).  A: [64,K] row-major,
// W: [N,K] row-major, out: [64,N] row-major.  Bias is mathematically absorbed
// by BN mean subtraction, so it is omitted (exact).
// Double-buffered LDS: chunk i+1 is prefetched into VGPRs (coalesced 8B
// loads; weights non-temporal since they stream once) while WMMAs consume
// chunk i; one barrier per chunk.
// flags: bit0 = affine (gamma/beta), bit1 = relu
// ---------------------------------------------------------------------------
__launch_bounds__(128)
__global__ void gemm_bn_kernel(const float* __restrict__ A,
                               const float* __restrict__ W,
                               const float* __restrict__ gamma,
                               const float* __restrict__ beta,
                               float* __restrict__ out,
                               int N, int K, int flags)
{
    __shared__ float Asl[2][MTOT * LDSS];
    __shared__ float Wsl[2][NTILE * LDSS];

    const int tid  = threadIdx.x;     // 0..127 (4 waves)
    const int wave = tid >> 5;        // 0..3  -> 16-column strip
    const int lane = tid & 31;
    const int lm   = lane & 15;       // row/col within 16
    const int kh   = (lane >> 4) << 1;// K sub-offset 0 or 2 (A/B frag layout)
    const int nblk = blockIdx.x * NTILE;
    const int nchunks = K / KC;

    v8f acc0 = {}, acc1 = {}, acc2 = {}, acc3 = {};
    v2f pa[8], pw[8];

    // Prologue: prefetch + stage chunk 0 into buffer 0.
    #pragma unroll
    for (int j = 0; j < 8; ++j) {
        const int q = tid + 128 * j;     // 1024 v2f per matrix per chunk
        const int r = q >> 4;            // 16 v2f per 32-float row
        const int c = (q & 15) << 1;
        pa[j] = *(const v2f*)&A[(size_t)r * K + c];
        pw[j] = __builtin_nontemporal_load((const v2f*)&W[(size_t)(nblk + r) * K + c]);
    }
    #pragma unroll
    for (int j = 0; j < 8; ++j) {
        const int q = tid + 128 * j;
        const int r = q >> 4;
        const int c = (q & 15) << 1;
        *(v2f*)&Asl[0][r * LDSS + c] = pa[j];
        *(v2f*)&Wsl[0][r * LDSS + c] = pw[j];
    }
    __syncthreads();

    for (int ch = 0; ch < nchunks; ++ch) {
        const int buf  = ch & 1;
        const bool more = (ch + 1) < nchunks;
        const int k0n  = (ch + 1) * KC;

        // Prefetch next chunk into registers (overlaps with WMMA below).
        if (more) {
            #pragma unroll
            for (int j = 0; j < 8; ++j) {
                const int q = tid + 128 * j;
                const int r = q >> 4;
                const int c = (q & 15) << 1;
                pa[j] = *(const v2f*)&A[(size_t)r * K + k0n + c];
                pw[j] = __builtin_nontemporal_load((const v2f*)&W[(size_t)(nblk + r) * K + k0n + c]);
            }
        }

        // Consume current chunk from LDS.
        const float* __restrict__ As = Asl[buf];
        const float* __restrict__ Ws = Wsl[buf];
        #pragma unroll
        for (int kk = 0; kk < KC; kk += 4) {
            v2f b  = *(const v2f*)&Ws[(wave * 16 + lm) * LDSS + kk + kh];
            v2f a0 = *(const v2f*)&As[(lm +  0) * LDSS + kk + kh];
            v2f a1 = *(const v2f*)&As[(lm + 16) * LDSS + kk + kh];
            v2f a2 = *(const v2f*)&As[(lm + 32) * LDSS + kk + kh];
            v2f a3 = *(const v2f*)&As[(lm + 48) * LDSS + kk + kh];
            acc0 = __builtin_amdgcn_wmma_f32_16x16x4_f32(false, a0, false, b, (short)0, acc0, false, false);
            acc1 = __builtin_amdgcn_wmma_f32_16x16x4_f32(false, a1, false, b, (short)0, acc1, false, false);
            acc2 = __builtin_amdgcn_wmma_f32_16x16x4_f32(false, a2, false, b, (short)0, acc2, false, false);
            acc3 = __builtin_amdgcn_wmma_f32_16x16x4_f32(false, a3, false, b, (short)0, acc3, false, false);
        }

        // Stage next chunk into the other buffer; single barrier per chunk.
        if (more) {
            float* __restrict__ Ad = Asl[buf ^ 1];
            float* __restrict__ Wd = Wsl[buf ^ 1];
            #pragma unroll
            for (int j = 0; j < 8; ++j) {
                const int q = tid + 128 * j;
                const int r = q >> 4;
                const int c = (q & 15) << 1;
                *(v2f*)&Ad[r * LDSS + c] = pa[j];
                *(v2f*)&Wd[r * LDSS + c] = pw[j];
            }
            __syncthreads();
        }
    }

    // ---- BatchNorm epilogue: stats over the 64 batch rows of this lane's
    // column.  Lane holds 32 rows; partner lane (xor 16) holds the other 32.
    float s = 0.f, sq = 0.f;
    #pragma unroll
    for (int r = 0; r < 8; ++r) {
        float x0 = acc0[r], x1 = acc1[r], x2 = acc2[r], x3 = acc3[r];
        s  += x0 + x1 + x2 + x3;
        sq += x0 * x0 + x1 * x1 + x2 * x2 + x3 * x3;
    }
    s  += __shfl_xor(s, 16, 32);
    sq += __shfl_xor(sq, 16, 32);

    const float mean = s * (1.f / 64.f);
    const float var  = sq * (1.f / 64.f) - mean * mean;
    const float rs   = rsqrtf(var + BN_EPS);

    const int ncol = nblk + wave * 16 + lm;
    float sa = rs, sb = -mean * rs;
    if (flags & 1) {
        const float g = gamma[ncol], bt = beta[ncol];
        sa = rs * g;
        sb = bt - mean * rs * g;
    }
    const bool do_relu = (flags & 2) != 0;
    const int rbase = (lane >> 4) ? 8 : 0;

    #pragma unroll
    for (int r = 0; r < 8; ++r) {
        float y0 = acc0[r] * sa + sb;
        float y1 = acc1[r] * sa + sb;
        float y2 = acc2[r] * sa + sb;
        float y3 = acc3[r] * sa + sb;
        if (do_relu) {
            y0 = fmaxf(y0, 0.f); y1 = fmaxf(y1, 0.f);
            y2 = fmaxf(y2, 0.f); y3 = fmaxf(y3, 0.f);
        }
        const int row = rbase + r;
        out[(size_t)(row +  0) * N + ncol] = y0;
        out[(size_t)(row + 16) * N + ncol] = y1;
        out[(size_t)(row + 32) * N + ncol] = y2;
        out[(size_t)(row + 48) * N + ncol] = y3;
    }
}

// ---------------------------------------------------------------------------
// JAX Threefry-2x32, key = PRNGKey(42) = (0, 42)
// ---------------------------------------------------------------------------
__device__ __forceinline__ unsigned rotl32(unsigned x, int r) {
    return (x << r) | (x >> (32 - r));
}

__device__ __forceinline__ void threefry2x32_42(unsigned x0, unsigned x1,
                                                unsigned& o0, unsigned& o1)
{
    const unsigned k0 = 0u, k1 = 42u;
    const unsigned k2 = 0x1BD11BDAu ^ k0 ^ k1;
    x0 += k0; x1 += k1;
#define TF_ROUND(r) { x0 += x1; x1 = rotl32(x1, (r)); x1 ^= x0; }
    TF_ROUND(13) TF_ROUND(15) TF_ROUND(26) TF_ROUND(6)
    x0 += k1; x1 += k2 + 1u;
    TF_ROUND(17) TF_ROUND(29) TF_ROUND(16) TF_ROUND(24)
    x0 += k2; x1 += k0 + 2u;
    TF_ROUND(13) TF_ROUND(15) TF_ROUND(26) TF_ROUND(6)
    x0 += k0; x1 += k1 + 3u;
    TF_ROUND(17) TF_ROUND(29) TF_ROUND(16) TF_ROUND(24)
    x0 += k1; x1 += k2 + 4u;
    TF_ROUND(13) TF_ROUND(15) TF_ROUND(26) TF_ROUND(6)
    x0 += k2; x1 += k0 + 5u;
#undef TF_ROUND
    o0 = x0; o1 = x1;
}

__device__ __forceinline__ float bits_to_gumbel(unsigned bits)
{
    // jax.random.uniform: u = bitcast((bits>>9)|0x3f800000) - 1, scaled to
    // [1e-10, 1).  gumbel = -log(-log(u))
    float uf = __uint_as_float((bits >> 9) | 0x3f800000u) - 1.0f;
    float u  = fmaxf(1e-10f, uf * (1.0f - 1e-10f) + 1e-10f);
    return -__logf(-__logf(u));
}

// ---------------------------------------------------------------------------
// Gumbel-softmax running max.
// grid = (32 k-chunks, 64 batch rows), block = 256 threads, 8 columns/thread.
// Chunk c handles samples k = 16c..16c+15 paired with k+512 (one Threefry
// call yields both halves of the flat (K,B,F) uniform tensor).
// ---------------------------------------------------------------------------
__device__ __forceinline__ float block_reduce_max(float v, float* red)
{
    #pragma unroll
    for (int off = 16; off > 0; off >>= 1) v = fmaxf(v, __shfl_xor(v, off, 32));
    const int tid = threadIdx.x;
    if ((tid & 31) == 0) red[tid >> 5] = v;
    __syncthreads();
    float r = red[0];
    #pragma unroll
    for (int w = 1; w < 8; ++w) r = fmaxf(r, red[w]);
    __syncthreads();
    return r;
}

__device__ __forceinline__ float block_reduce_sum(float v, float* red)
{
    #pragma unroll
    for (int off = 16; off > 0; off >>= 1) v += __shfl_xor(v, off, 32);
    const int tid = threadIdx.x;
    if ((tid & 31) == 0) red[tid >> 5] = v;
    __syncthreads();
    float r = red[0];
    #pragma unroll
    for (int w = 1; w < 8; ++w) r += red[w];
    __syncthreads();
    return r;
}

__launch_bounds__(256)
__global__ void gumbel_softmax_max_kernel(const float* __restrict__ logits, // [64,2048]
                                          unsigned* __restrict__ zbits)     // [64,2048] f32 bits
{
    __shared__ float red[8];
    const int tid = threadIdx.x;
    const int c   = blockIdx.x;   // 0..31
    const int b   = blockIdx.y;   // 0..63

    float lg[8], zacc[8];
    #pragma unroll
    for (int j = 0; j < 8; ++j) {
        lg[j]   = logits[b * 2048 + tid + 256 * j];
        zacc[j] = 0.f;
    }

    for (int kp = 0; kp < 16; ++kp) {
        const int k0 = c * 16 + kp;          // sample in [0,512)
        float g0[8], g1[8];
        #pragma unroll
        for (int j = 0; j < 8; ++j) {
            const int f = tid + 256 * j;
            // flat index into (K=1024, B=64, F=2048); counter pair (i, i+2^26)
            const unsigned i = ((unsigned)(k0 * 64 + b) << 11) + (unsigned)f;
            unsigned r0, r1;
            threefry2x32_42(i, i + (1u << 26), r0, r1);
            g0[j] = bits_to_gumbel(r0);      // sample k0
            g1[j] = bits_to_gumbel(r1);      // sample k0 + 512
        }
        #pragma unroll
        for (int half = 0; half < 2; ++half) {
            float t[8];
            float m = -3.4e38f;
            #pragma unroll
            for (int j = 0; j < 8; ++j) {
                const float g = half ? g1[j] : g0[j];
                t[j] = (lg[j] + g) * 2.0f;   // / tau, tau = 0.5
                m = fmaxf(m, t[j]);
            }
            m = block_reduce_max(m, red);
            float s = 0.f, e[8];
            #pragma unroll
            for (int j = 0; j < 8; ++j) { e[j] = __expf(t[j] - m); s += e[j]; }
            s = block_reduce_sum(s, red);
            const float inv = 1.0f / s;
            #pragma unroll
            for (int j = 0; j < 8; ++j) zacc[j] = fmaxf(zacc[j], e[j] * inv);
        }
    }

    // Running max over K across blocks: softmax outputs are positive, so
    // uint-bit atomicMax == float max (d_out zero-initialized).
    #pragma unroll
    for (int j = 0; j < 8; ++j)
        atomicMax(&zbits[b * 2048 + tid + 256 * j], __float_as_uint(zacc[j]));
}

__global__ void zero_kernel(float* __restrict__ p, int n)
{
    const int i = blockIdx.x * 256 + threadIdx.x;
    if (i < n) p[i] = 0.f;
}

// ---------------------------------------------------------------------------
extern "C" void kernel_launch(void* const* d_in, const int* in_sizes, int n_in,
                              void* d_out, int out_size, void* d_ws, size_t ws_size,
                              hipStream_t stream)
{
    (void)in_sizes; (void)n_in; (void)out_size; (void)ws_size;

    const float* f      = (const float*)d_in[0];
    const float* W1     = (const float*)d_in[1];
    const float* gamma1 = (const float*)d_in[3];
    const float* beta1  = (const float*)d_in[4];
    const float* W2     = (const float*)d_in[5];
    const float* gamma2 = (const float*)d_in[7];
    const float* beta2  = (const float*)d_in[8];
    const float* W3     = (const float*)d_in[9];
    // b1/b2/b3 (d_in[2], d_in[6], d_in[10]) are exactly absorbed by BN.

    float* h1     = (float*)d_ws;                 // [64, 8192]
    float* h2     = h1 + (size_t)64 * 8192;       // [64, 8192]
    float* logits = h2 + (size_t)64 * 8192;       // [64, 2048]
    float* z      = (float*)d_out;                // [64, 2048]

    // Layer 1: [64,2048] @ [8192,2048]^T -> BN affine + ReLU
    gemm_bn_kernel<<<8192 / NTILE, 128, 0, stream>>>(f,  W1, gamma1, beta1, h1, 8192, 2048, 3);
    // Layer 2: [64,8192] @ [8192,8192]^T -> BN affine + ReLU
    gemm_bn_kernel<<<8192 / NTILE, 128, 0, stream>>>(h1, W2, gamma2, beta2, h2, 8192, 8192, 3);
    // Layer 3: [64,8192] @ [2048,8192]^T -> BN (no affine, no relu)
    gemm_bn_kernel<<<2048 / NTILE, 128, 0, stream>>>(h2, W3, nullptr, nullptr, logits, 2048, 8192, 0);

    // z = max_k softmax((logits + gumbel_k)/tau)
    zero_kernel<<<(64 * 2048 + 255) / 256, 256, 0, stream>>>(z, 64 * 2048);
    gumbel_softmax_max_kernel<<<dim3(32, 64), 256, 0, stream>>>(logits, (unsigned*)z);
}